// GIN_87101936762929
// MI455X (gfx1250) — compile-verified
//
#include <hip/hip_runtime.h>
#include <hip/hip_bf16.h>

#define HID 256
#define LDSS 260  // padded LDS row stride (floats) -> conflict-free WMMA frag reads

typedef float v2f __attribute__((ext_vector_type(2)));
typedef float v8f __attribute__((ext_vector_type(8)));

#if defined(__has_builtin)
#if __has_builtin(__builtin_amdgcn_global_load_async_to_lds_b128)
#define HAVE_ASYNC_LDS 1
#endif
#if __has_builtin(__builtin_amdgcn_s_wait_asynccnt)
#define HAVE_WAIT_ASYNC 1
#endif
#endif

#ifdef HAVE_ASYNC_LDS
typedef int b128_t __attribute__((vector_size(16)));
typedef __attribute__((address_space(1))) b128_t* glb128_t;
typedef __attribute__((address_space(3))) b128_t* ldsb128_t;
#endif

// hardware fp32 atomic add (global_atomic_add_f32, no CAS loop)
__device__ __forceinline__ void atomAddF32(float* p, float v) {
#if defined(__AMDGCN__)
    unsafeAtomicAdd(p, v);
#else
    atomicAdd(p, v);
#endif
}

// ---------------- zero ----------------
__global__ void gin_zero_kernel(float* __restrict__ p, int n) {
    int i = blockIdx.x * 256 + threadIdx.x;
    if (i < n) p[i] = 0.0f;
}

// ---------------- input embeddings: z = xe1[x0] + xe2[x1] ----------------
__global__ void gin_embed_kernel(const int* __restrict__ x,
                                 const float* __restrict__ xe1,
                                 const float* __restrict__ xe2,
                                 float* __restrict__ z, int nNodes) {
    int idx = blockIdx.x * 256 + threadIdx.x;
    int node = idx >> 6;           // 64 float4 chunks per row
    int c4 = (idx & 63) * 4;
    if (node >= nNodes) return;
    int t0 = x[2 * node], t1 = x[2 * node + 1];
    float4 a = *(const float4*)(xe1 + (size_t)t0 * HID + c4);
    float4 b = *(const float4*)(xe2 + (size_t)t1 * HID + c4);
    float4 o = make_float4(a.x + b.x, a.y + b.y, a.z + b.z, a.w + b.w);
    *(float4*)(z + (size_t)node * HID + c4) = o;
}

// ---------------- combined bond embedding table: ecomb[a0*3+a1] = ee1[a0]+ee2[a1]
__global__ void gin_ecomb_kernel(const float* __restrict__ ee1,
                                 const float* __restrict__ ee2,
                                 float* __restrict__ ecomb) {
    int row = blockIdx.x;          // 18 rows (6 bond types x 3 dirs)
    int f = threadIdx.x;           // 256
    ecomb[row * HID + f] = ee1[(row / 3) * HID + f] + ee2[(row % 3) * HID + f];
}

// ---------------- agg = z (so scatter produces (1+eps)*z + sum) ----------------
__global__ void gin_copy_kernel(const float4* __restrict__ src,
                                float4* __restrict__ dst, int n4) {
    int i = blockIdx.x * 256 + threadIdx.x;
    if (i < n4) dst[i] = src[i];
}

// ---------------- edge scatter: agg[dst] += relu(z[src] + ecomb[attr]) --------
__global__ __launch_bounds__(256) void gin_edge_kernel(
    const float* __restrict__ z, const float* __restrict__ ecomb,
    const int* __restrict__ ei, const int* __restrict__ ea,
    float* __restrict__ agg, int nEdges) {
    int e = blockIdx.x * 8 + (threadIdx.x >> 5);   // one wave per edge
    if (e >= nEdges) return;
    int lane = threadIdx.x & 31;
    int src = ei[e];
    int dst = ei[nEdges + e];
    int a0 = ea[2 * e], a1 = ea[2 * e + 1];
    const float* zr = z + (size_t)src * HID;
    const float* er = ecomb + (size_t)(a0 * 3 + a1) * HID;
    float* ar = agg + (size_t)dst * HID;
#pragma unroll
    for (int half = 0; half < 2; ++half) {
        int c = half * 128 + lane * 4;
        float4 zv = *(const float4*)(zr + c);
        float4 ev = *(const float4*)(er + c);
        atomAddF32(ar + c + 0, fmaxf(zv.x + ev.x, 0.0f));
        atomAddF32(ar + c + 1, fmaxf(zv.y + ev.y, 0.0f));
        atomAddF32(ar + c + 2, fmaxf(zv.z + ev.z, 0.0f));
        atomAddF32(ar + c + 3, fmaxf(zv.w + ev.w, 0.0f));
    }
}

// ---------------- fused MLP: z = relu(relu(h0@W1+b1)@W2+b2), WMMA f32 --------
// Block: 256 threads = 8 wave32, owns 16 rows x 256 cols. Each wave: two 16x16
// output tiles. A frag per ISA: lanes0-15 rowM, K={k,k+1}; lanes16-31 K={k+2,k+3}.
// B frag: VGPR v holds K rows {k+v (lanes0-15), k+v+2 (lanes16-31)}, N=lane&15.
__global__ __launch_bounds__(256) void gin_mlp_kernel(
    const float* __restrict__ h0, const float* __restrict__ W1,
    const float* __restrict__ bias1, const float* __restrict__ W2,
    const float* __restrict__ bias2, float* __restrict__ zout, int nNodes) {
    __shared__ __align__(16) float sA[16 * LDSS];
    __shared__ __align__(16) float sH[16 * LDSS];
    const int tid = threadIdx.x;
    const int lane = tid & 31;
    const int wave = tid >> 5;
    const int m0 = blockIdx.x * 16;

    // stage h0 tile into LDS
    if (m0 + 16 <= nNodes) {
#ifdef HAVE_ASYNC_LDS
        // CDNA5 async global->LDS (bypasses VGPRs, tracked on ASYNCcnt):
        // 16 rows x 64 float4 = 1024 b128 items, 256 lanes -> 4 per lane.
#pragma unroll
        for (int it = 0; it < 4; ++it) {
            int i = it * 256 + tid;
            int r = i >> 6;
            int c4 = (i & 63) * 4;
            __builtin_amdgcn_global_load_async_to_lds_b128(
                (glb128_t)(h0 + (size_t)(m0 + r) * HID + c4),
                (ldsb128_t)&sA[r * LDSS + c4],
                0, 0);
        }
#ifdef HAVE_WAIT_ASYNC
        __builtin_amdgcn_s_wait_asynccnt(0);
#else
        asm volatile("s_wait_asynccnt 0x0" ::: "memory");
#endif
#else
        for (int i = tid; i < 16 * HID; i += 256) {
            int r = i >> 8, c = i & 255;
            sA[r * LDSS + c] = h0[(size_t)(m0 + r) * HID + c];
        }
#endif
    } else {  // tail tile (not taken when N % 16 == 0): guarded, zero-padded
        for (int i = tid; i < 16 * HID; i += 256) {
            int r = i >> 8, c = i & 255;
            int row = m0 + r;
            sA[r * LDSS + c] = (row < nNodes) ? h0[(size_t)row * HID + c] : 0.0f;
        }
    }
    __syncthreads();

    const int hi = lane >> 4;       // lane half selects K pair
    const int ml = lane & 15;
    const int n0 = wave * 32;       // 8 waves x 32 cols = 256

    // ---- GEMM1: sA @ W1 ----
    v8f c0 = {}; v8f c1 = {};
#pragma unroll 4
    for (int k = 0; k < HID; k += 4) {
        int ka = k + 2 * hi;
        v2f a; a.x = sA[ml * LDSS + ka]; a.y = sA[ml * LDSS + ka + 1];
        const float* Wp = W1 + (size_t)ka * HID;
        v2f b0; b0.x = Wp[n0 + ml];        b0.y = Wp[HID + n0 + ml];
        v2f b1; b1.x = Wp[n0 + 16 + ml];   b1.y = Wp[HID + n0 + 16 + ml];
        c0 = __builtin_amdgcn_wmma_f32_16x16x4_f32(false, a, false, b0,
                                                   (short)0, c0, false, false);
        c1 = __builtin_amdgcn_wmma_f32_16x16x4_f32(false, a, false, b1,
                                                   (short)0, c1, false, false);
    }
    {
        float bv0 = bias1[n0 + ml];
        float bv1 = bias1[n0 + 16 + ml];
#pragma unroll
        for (int r = 0; r < 8; ++r) {
            int M = r + 8 * hi;            // C/D layout: VGPR r -> rows r / r+8
            sH[M * LDSS + n0 + ml]      = fmaxf(c0[r] + bv0, 0.0f);
            sH[M * LDSS + n0 + 16 + ml] = fmaxf(c1[r] + bv1, 0.0f);
        }
    }
    __syncthreads();

    // ---- GEMM2: sH @ W2 ----
    c0 = (v8f){}; c1 = (v8f){};
#pragma unroll 4
    for (int k = 0; k < HID; k += 4) {
        int ka = k + 2 * hi;
        v2f a; a.x = sH[ml * LDSS + ka]; a.y = sH[ml * LDSS + ka + 1];
        const float* Wp = W2 + (size_t)ka * HID;
        v2f b0; b0.x = Wp[n0 + ml];        b0.y = Wp[HID + n0 + ml];
        v2f b1; b1.x = Wp[n0 + 16 + ml];   b1.y = Wp[HID + n0 + 16 + ml];
        c0 = __builtin_amdgcn_wmma_f32_16x16x4_f32(false, a, false, b0,
                                                   (short)0, c0, false, false);
        c1 = __builtin_amdgcn_wmma_f32_16x16x4_f32(false, a, false, b1,
                                                   (short)0, c1, false, false);
    }
    {
        float bv0 = bias2[n0 + ml];
        float bv1 = bias2[n0 + 16 + ml];
#pragma unroll
        for (int r = 0; r < 8; ++r) {
            int M = r + 8 * hi;
            int row = m0 + M;
            if (row < nNodes) {
                zout[(size_t)row * HID + n0 + ml]      = fmaxf(c0[r] + bv0, 0.0f);
                zout[(size_t)row * HID + n0 + 16 + ml] = fmaxf(c1[r] + bv1, 0.0f);
            }
        }
    }
}

// ---------------- global mean pool: sums + counts ----------------
__global__ void gin_pool_kernel(const float* __restrict__ z,
                                const int* __restrict__ batch,
                                float* __restrict__ gsum,
                                float* __restrict__ cnt, int nNodes) {
    int idx = blockIdx.x * 256 + threadIdx.x;
    int node = idx >> 6;
    int c4 = (idx & 63) * 4;
    if (node >= nNodes) return;
    int g = batch[node];
    float4 v = *(const float4*)(z + (size_t)node * HID + c4);
    float* gr = gsum + (size_t)g * HID + c4;
    atomAddF32(gr + 0, v.x);
    atomAddF32(gr + 1, v.y);
    atomAddF32(gr + 2, v.z);
    atomAddF32(gr + 3, v.w);
    if (c4 == 0) atomAddF32(cnt + g, 1.0f);
}

__global__ void gin_pooldiv_kernel(float* __restrict__ g,
                                   const float* __restrict__ cnt, int n) {
    int idx = blockIdx.x * 256 + threadIdx.x;
    if (idx >= n) return;
    float c = fmaxf(cnt[idx >> 8], 1.0f);
    g[idx] = g[idx] / c;
}

extern "C" void kernel_launch(void* const* d_in, const int* in_sizes, int n_in,
                              void* d_out, int out_size, void* d_ws, size_t ws_size,
                              hipStream_t stream) {
    const int*   x     = (const int*)d_in[0];
    const int*   ei    = (const int*)d_in[1];
    const int*   ea    = (const int*)d_in[2];
    const int*   batch = (const int*)d_in[3];
    const float* xe1   = (const float*)d_in[4];
    const float* xe2   = (const float*)d_in[5];
    const float* ee1   = (const float*)d_in[6];
    const float* ee2   = (const float*)d_in[7];
    const float* W1    = (const float*)d_in[8];
    const float* b1    = (const float*)d_in[9];
    const float* W2    = (const float*)d_in[10];
    const float* b2    = (const float*)d_in[11];

    const int nNodes  = in_sizes[3];
    const int nEdges  = in_sizes[1] / 2;
    const int nLayers = in_sizes[9] / HID;
    const int nGraphs = (out_size - nNodes * HID) / HID;

    float* out_z = (float*)d_out;                        // [N,256]
    float* out_g = (float*)d_out + (size_t)nNodes * HID; // [G,256]

    char* ws = (char*)d_ws;
    size_t zbytes = (size_t)nNodes * HID * sizeof(float);
    float* zbuf  = (float*)(ws);
    float* agg   = (float*)(ws + zbytes);
    float* ecomb = (float*)(ws + 2 * zbytes);
    float* cnt   = (float*)(ws + 2 * zbytes + 18 * HID * sizeof(float));

    const int featBlocks = (nNodes * 64 + 255) / 256;    // node*64 float4 items
    const int n4 = nNodes * 64;

    // input embeddings + combined edge table
    gin_embed_kernel<<<featBlocks, 256, 0, stream>>>(x, xe1, xe2, zbuf, nNodes);
    gin_ecomb_kernel<<<18, 256, 0, stream>>>(ee1, ee2, ecomb);

    const int edgeBlocks = (nEdges + 7) / 8;
    const int mlpBlocks  = (nNodes + 15) / 16;

    for (int l = 0; l < nLayers; ++l) {
        gin_copy_kernel<<<(n4 + 255) / 256, 256, 0, stream>>>(
            (const float4*)zbuf, (float4*)agg, n4);
        gin_edge_kernel<<<edgeBlocks, 256, 0, stream>>>(
            zbuf, ecomb, ei, ea, agg, nEdges);
        float* zo = (l == nLayers - 1) ? out_z : zbuf;
        gin_mlp_kernel<<<mlpBlocks, 256, 0, stream>>>(
            agg, W1 + (size_t)l * HID * HID, b1 + (size_t)l * HID,
            W2 + (size_t)l * HID * HID, b2 + (size_t)l * HID, zo, nNodes);
    }

    // mean pool
    int gElems = nGraphs * HID;
    gin_zero_kernel<<<(gElems + 255) / 256, 256, 0, stream>>>(out_g, gElems);
    gin_zero_kernel<<<1, 256, 0, stream>>>(cnt, nGraphs);
    gin_pool_kernel<<<featBlocks, 256, 0, stream>>>(out_z, batch, out_g, cnt, nNodes);
    gin_pooldiv_kernel<<<(gElems + 255) / 256, 256, 0, stream>>>(out_g, cnt, gElems);
}